// SegDecoder_25924422598893
// MI455X (gfx1250) — compile-verified
//
#include <hip/hip_runtime.h>

// ---------------------------------------------------------------------------
// SegDecoder for MI455X (gfx1250, wave32, WMMA).
//
// Pipeline:
//   1..3) gemm_wmma_f32 : x = [gelu](x @ W.T + b)   (512x1024 x 1024x1024)
//   4)    gemm_wmma_f32 : logits[t] = seg_h[:,t,:] @ v_h[t].T   (16 batches)
//   5)    resize448     : bilinear 32x32 -> 448x448 (half-pixel, edge clamp)
//
// The 411 MB fp32 output write dominates (~18 us at 23.3 TB/s HBM); GEMMs use
// V_WMMA_F32_16X16X4_F32 to keep fp32 precision while staying off the slow
// vector-FMA path.
// ---------------------------------------------------------------------------

typedef float v2f __attribute__((ext_vector_type(2)));
typedef float v8f __attribute__((ext_vector_type(8)));

#define TM 32   // tile rows (M)
#define TN 64   // tile cols (N)
#define KC 32   // K chunk staged in LDS
#define SA 36   // padded LDS row stride in floats (36 ~ conflict-free frags,
                // multiple of 4 so rows stay 16B-aligned for b128 stores)

__device__ __forceinline__ float gelu_erf(float x) {
  // exact (erf) GELU, matches jax.nn.gelu(approximate=False)
  return 0.5f * x * (1.0f + erff(x * 0.70710678118654752f));
}

__device__ __forceinline__ int clampi(int v, int lo, int hi) {
  return v < lo ? lo : (v > hi ? hi : v);
}

// D[m,n] = sum_k A[m,k] * B[n,k]  (+bias[n]) (+gelu), fp32 via WMMA.
// Per-batch (blockIdx.z) element offsets allow the 16-way einsum reuse.
__global__ __launch_bounds__(128)
void gemm_wmma_f32(const float* __restrict__ A, long long lda, long long aBatch,
                   const float* __restrict__ B, long long ldb, long long bBatch,
                   const float* __restrict__ bias,
                   float* __restrict__ C, long long ldc, long long cBatch,
                   int K, int doGelu)
{
  __shared__ float As[TM * SA];
  __shared__ float Bs[TN * SA];

  const int tid  = threadIdx.x;
  const int lane = tid & 31;
  const int wave = tid >> 5;          // 4 waves
  const int lm   = lane & 15;         // row-in-tile for A/B frags, col for D
  const int lkh  = (lane >> 4) << 1;  // k sub-offset: 0 (lanes 0-15) / 2 (16-31)
  const int mrow = (wave & 1) << 4;   // 0 / 16 : wave's M strip inside tile
  const int ncol = (wave >> 1) << 5;  // 0 / 32 : wave's pair of N tiles

  const size_t za = (size_t)blockIdx.z * (size_t)aBatch;
  const size_t zb = (size_t)blockIdx.z * (size_t)bBatch;
  const size_t zc = (size_t)blockIdx.z * (size_t)cBatch;
  const int m0 = blockIdx.y * TM;
  const int n0 = blockIdx.x * TN;

  v8f acc0 = {};
  v8f acc1 = {};

  for (int kt = 0; kt < K; kt += KC) {
    // --- cooperative global -> LDS, float4 (global_load_b128) per op -------
    #pragma unroll
    for (int i = 0; i < (TM * KC / 4) / 128; ++i) {       // 2 iters
      int idx = tid + i * 128;
      int r   = idx >> 3;                                  // KC/4 == 8
      int c4  = (idx & 7) << 2;
      const float4 v = *reinterpret_cast<const float4*>(
          A + za + (size_t)(m0 + r) * (size_t)lda + (size_t)(kt + c4));
      *reinterpret_cast<float4*>(&As[r * SA + c4]) = v;
    }
    #pragma unroll
    for (int i = 0; i < (TN * KC / 4) / 128; ++i) {       // 4 iters
      int idx = tid + i * 128;
      int r   = idx >> 3;
      int c4  = (idx & 7) << 2;
      const float4 v = *reinterpret_cast<const float4*>(
          B + zb + (size_t)(n0 + r) * (size_t)ldb + (size_t)(kt + c4));
      *reinterpret_cast<float4*>(&Bs[r * SA + c4]) = v;
    }
    __syncthreads();

    // --- 8 WMMA k-steps per chunk, 2 N-tiles per wave ----------------------
    // A frag (16x4 f32): lane L -> A[lm][k0 + lkh + {0,1}]
    // B frag (4x16, =W^T): lane L -> W[n=lm][k0 + lkh + {0,1}] (same pattern)
    #pragma unroll
    for (int k0 = 0; k0 < KC; k0 += 4) {
      v2f a  = *reinterpret_cast<const v2f*>(&As[(mrow + lm) * SA + k0 + lkh]);
      v2f b0 = *reinterpret_cast<const v2f*>(&Bs[(ncol + lm) * SA + k0 + lkh]);
      v2f b1 = *reinterpret_cast<const v2f*>(&Bs[(ncol + 16 + lm) * SA + k0 + lkh]);
      acc0 = __builtin_amdgcn_wmma_f32_16x16x4_f32(false, a, false, b0,
                                                   (short)0, acc0, false, false);
      acc1 = __builtin_amdgcn_wmma_f32_16x16x4_f32(false, a, false, b1,
                                                   (short)0, acc1, false, false);
    }
    __syncthreads();
  }

  // --- epilogue: D VGPR v -> row (v + 8*(lane>>4)), col (lane&15) ----------
  const int gm  = m0 + mrow + ((lane >> 4) << 3);
  const int gn0 = n0 + ncol + lm;
  const int gn1 = gn0 + 16;
  const float bv0 = bias ? bias[gn0] : 0.0f;
  const float bv1 = bias ? bias[gn1] : 0.0f;
  #pragma unroll
  for (int v = 0; v < 8; ++v) {
    float o0 = acc0[v] + bv0;
    float o1 = acc1[v] + bv1;
    if (doGelu) { o0 = gelu_erf(o0); o1 = gelu_erf(o1); }
    C[zc + (size_t)(gm + v) * (size_t)ldc + gn0] = o0;
    C[zc + (size_t)(gm + v) * (size_t)ldc + gn1] = o1;
  }
}

// Bilinear 32x32 -> 448x448, half-pixel centers, edge clamp.
// grid = (512 images, 448 rows), block = 224 threads (7 waves), 2 px/thread.
__global__ __launch_bounds__(224)
void resize448(const float* __restrict__ src_all, float* __restrict__ dst_all)
{
  const int img = blockIdx.x;          // n*16 + t
  const int y   = blockIdx.y;          // 0..447
  const float* __restrict__ src = src_all + (size_t)img * 1024;
  float* __restrict__ dst = dst_all + (size_t)img * (448 * 448) + (size_t)y * 448;

  const float sy  = (y + 0.5f) * (1.0f / 14.0f) - 0.5f;
  const float y0f = floorf(sy);
  const float fy  = sy - y0f;
  const int   yi  = (int)y0f;
  const float* __restrict__ r0 = src + clampi(yi,     0, 31) * 32;
  const float* __restrict__ r1 = src + clampi(yi + 1, 0, 31) * 32;

  const int xbase = threadIdx.x * 2;
  float o[2];
  #pragma unroll
  for (int j = 0; j < 2; ++j) {
    const int   x   = xbase + j;
    const float sx  = (x + 0.5f) * (1.0f / 14.0f) - 0.5f;
    const float x0f = floorf(sx);
    const float fx  = sx - x0f;
    const int   xi  = (int)x0f;
    const int   x0  = clampi(xi,     0, 31);
    const int   x1  = clampi(xi + 1, 0, 31);
    const float top = r0[x0] + fx * (r0[x1] - r0[x0]);
    const float bot = r1[x0] + fx * (r1[x1] - r1[x0]);
    o[j] = top + fy * (bot - top);
  }
  *reinterpret_cast<float2*>(&dst[xbase]) = make_float2(o[0], o[1]);
}

extern "C" void kernel_launch(void* const* d_in, const int* in_sizes, int n_in,
                              void* d_out, int out_size, void* d_ws, size_t ws_size,
                              hipStream_t stream)
{
  (void)in_sizes; (void)n_in; (void)out_size; (void)ws_size;
  const float* seg = (const float*)d_in[0];   // [32,16,1024]
  const float* vis = (const float*)d_in[1];   // [1,16384,1024]
  // d_in[2] = grid_thw (int64), shape-only: unused
  const float* W1  = (const float*)d_in[3];
  const float* b1  = (const float*)d_in[4];
  const float* W2  = (const float*)d_in[5];
  const float* b2  = (const float*)d_in[6];
  const float* W3  = (const float*)d_in[7];
  const float* b3  = (const float*)d_in[8];
  float* out = (float*)d_out;                 // [32,16,448,448]

  const size_t SZ = 512 * 1024;               // one [512,1024] fp32 buffer
  float* x1 = (float*)d_ws;                   // gelu(seg@W1.T+b1)
  float* x2 = x1 + SZ;                        // gelu(x1@W2.T+b2)
  float* sh = x2 + SZ;                        // seg_h = x2@W3.T+b3
  float* lg = sh + SZ;                        // logits [32,16,32,32] (=512x1024)

  const dim3 blk(128);
  const dim3 gMLP(1024 / TN, 512 / TM, 1);    // 16 x 16 workgroups
  const int C = 1024;

  // MLP: A rows are the flattened [N*T, C] activations; B rows are W rows.
  gemm_wmma_f32<<<gMLP, blk, 0, stream>>>(seg, C, 0, W1, C, 0, b1, x1, C, 0, C, 1);
  gemm_wmma_f32<<<gMLP, blk, 0, stream>>>(x1,  C, 0, W2, C, 0, b2, x2, C, 0, C, 1);
  gemm_wmma_f32<<<gMLP, blk, 0, stream>>>(x2,  C, 0, W3, C, 0, b3, sh, C, 0, C, 0);

  // Einsum 'ntc,thwc->nthw' as 16 batched GEMMs (z = t):
  //   A row n  : sh[n*16384 + t*1024 + k]      -> lda 16384, aBatch 1024
  //   B row p  : vis[(t*1024+p)*1024 + k]      -> ldb 1024,  bBatch 1048576
  //   C row n  : lg[n*16384 + t*1024 + p]      -> ldc 16384, cBatch 1024
  const dim3 gEIN(1024 / TN, 32 / TM, 16);
  gemm_wmma_f32<<<gEIN, blk, 0, stream>>>(sh, 16384, 1024,
                                          vis, 1024, 1024LL * 1024,
                                          nullptr,
                                          lg, 16384, 1024, C, 0);

  // Bilinear upsample: one block per (image, output row).
  resize448<<<dim3(512, 448, 1), dim3(224), 0, stream>>>(lg, out);
}